// Attention_spd_35038343200973
// MI455X (gfx1250) — compile-verified
//
#include <hip/hip_runtime.h>

// Attention with shared positional bias (spd), head-keep rescale, QKV + output
// projections. MI455X (gfx1250, wave32) implementation:
//   K1: qkv = x @ w_qkv           -> q/k/v in [b][h][n][64] layout (ws)
//   K2: flash attention (spd bias read exactly once, scores never hit memory)
//   K3: out = attn_out @ w_out + b_out
// All matrix math via V_WMMA_F32_16X16X4_F32 (f32-native CDNA5 WMMA).
// Workspace use: 4 * 8MB = 32 MB (q, k, v, attn_out).

#define BATCH 4
#define SEQ   1024
#define DIM   512
#define HEADS 8
#define DHEAD 64
#define NQKV  1536
#define QKVSZ (BATCH * HEADS * SEQ * DHEAD)  // 2,097,152 floats per tensor

typedef float v2f __attribute__((ext_vector_type(2)));
typedef float v8f __attribute__((ext_vector_type(8)));

__device__ __forceinline__ v8f wmma4(v2f a, v2f b, v8f c) {
  // D = A(16x4 f32) * B(4x16 f32) + C(16x16 f32)
  return __builtin_amdgcn_wmma_f32_16x16x4_f32(
      /*neg_a=*/false, a, /*neg_b=*/false, b,
      /*c_mod=*/(short)0, c, /*reuse_a=*/false, /*reuse_b=*/false);
}

// ---------------------------------------------------------------------------
// K1: qkv GEMM.  X[4096][512] @ W[512][1536] -> q/k/v each [b][h][n][64].
// One wave per workgroup; wave tile = 32(M) x 64(N); K loop step 4.
// ---------------------------------------------------------------------------
__global__ void __launch_bounds__(32) qkv_gemm_kernel(
    const float* __restrict__ X, const float* __restrict__ W,
    float* __restrict__ qkv) {
  const int lane = threadIdx.x & 31;
  const int half = lane >> 4;   // 0: K pair {0,1}, 1: K pair {2,3}
  const int l16  = lane & 15;
  const int m0 = blockIdx.x * 32;
  const int n0 = blockIdx.y * 64;

  const v8f zero8 = {0.f, 0.f, 0.f, 0.f, 0.f, 0.f, 0.f, 0.f};
  v8f acc[2][4];
#pragma unroll
  for (int mi = 0; mi < 2; ++mi)
#pragma unroll
    for (int nb = 0; nb < 4; ++nb) acc[mi][nb] = zero8;

  const float* Arow0 = X + (size_t)(m0 + l16) * DIM;
  const float* Arow1 = X + (size_t)(m0 + 16 + l16) * DIM;

  for (int k0 = 0; k0 < DIM; k0 += 4) {
    const int ka = k0 + 2 * half;
    v2f a0 = *(const v2f*)(Arow0 + ka);
    v2f a1 = *(const v2f*)(Arow1 + ka);
#pragma unroll
    for (int nb = 0; nb < 4; ++nb) {
      const int n = n0 + nb * 16 + l16;
      v2f b;
      b.x = W[(size_t)(ka    ) * NQKV + n];
      b.y = W[(size_t)(ka + 1) * NQKV + n];
      acc[0][nb] = wmma4(a0, b, acc[0][nb]);
      acc[1][nb] = wmma4(a1, b, acc[1][nb]);
    }
  }

  // Scatter into q/k/v head-major layout: seg*[b][h][n][d]
#pragma unroll
  for (int mi = 0; mi < 2; ++mi) {
#pragma unroll
    for (int nb = 0; nb < 4; ++nb) {
      const int col = n0 + nb * 16 + l16;
      const int seg = col >> 9;         // 0=q 1=k 2=v
      const int rem = col & 511;
      const int h   = rem >> 6;
      const int d   = rem & 63;
      float* dst = qkv + (size_t)seg * QKVSZ;
#pragma unroll
      for (int r = 0; r < 8; ++r) {
        const int grow = m0 + mi * 16 + (half << 3) + r;  // 0..4095
        const int bb   = grow >> 10;
        const int srow = grow & 1023;
        dst[((size_t)(bb * HEADS + h) * SEQ + srow) * DHEAD + d] =
            acc[mi][nb][r];
      }
    }
  }
}

// ---------------------------------------------------------------------------
// K2: flash attention.  Per (b,h): softmax(scale*Q@K^T + spd) @ V, with
// head_keep rescale.  Block = 128 threads (4 waves), each wave owns 16 query
// rows; inner loop over 64-key blocks staged in LDS (K row-major, V
// transposed, stride 68 floats for bank-conflict-free float2 fragment reads).
// ---------------------------------------------------------------------------
__global__ void __launch_bounds__(128) flash_attn_kernel(
    const float* __restrict__ Q, const float* __restrict__ K,
    const float* __restrict__ V, const float* __restrict__ SPD,
    const float* __restrict__ head_keep, float* __restrict__ AO) {
  __shared__ float Kt[64 * 68];      // [key][dim]
  __shared__ float Vt[64 * 68];      // [dim][key] (transposed)
  __shared__ float Pw[4 * 16 * 68];  // per-wave probability tile [row][key]

  const int tid  = threadIdx.x;
  const int wave = tid >> 5;
  const int lane = tid & 31;
  const int half = lane >> 4;
  const int l16  = lane & 15;

  const int qb = blockIdx.x;  // 16 query blocks of 64
  const int h  = blockIdx.y;  // 8 heads
  const int b  = blockIdx.z;  // 4 batches
  const int bh = b * HEADS + h;
  const int q0 = qb * 64 + wave * 16;  // this wave's first query row

  const float* Qp = Q + ((size_t)bh * SEQ + q0) * DHEAD;
  const float* Kp = K + (size_t)bh * SEQ * DHEAD;
  const float* Vp = V + (size_t)bh * SEQ * DHEAD;
  const float* Sp = SPD + (size_t)(bh * SEQ + q0) * SEQ;

  const float SCALE = 0.125f;  // 64^-0.5

  // Preload Q as 16 A-fragments (pre-scaled): kbase = 4*kk + 2*half.
  v2f qf[16];
#pragma unroll
  for (int kk = 0; kk < 16; ++kk) {
    v2f t = *(const v2f*)(Qp + (size_t)l16 * DHEAD + 4 * kk + 2 * half);
    qf[kk].x = t.x * SCALE;
    qf[kk].y = t.y * SCALE;
  }

  const v8f zero8 = {0.f, 0.f, 0.f, 0.f, 0.f, 0.f, 0.f, 0.f};
  v8f o[4];
#pragma unroll
  for (int db = 0; db < 4; ++db) o[db] = zero8;
  float mrow[8], lrow[8];
#pragma unroll
  for (int r = 0; r < 8; ++r) { mrow[r] = -1.0e30f; lrow[r] = 0.f; }

  float* Pl = Pw + wave * (16 * 68);

  for (int j0 = 0; j0 < SEQ; j0 += 64) {
    __syncthreads();  // previous block's Kt/Vt reads complete
    // Cooperative stage of 64x64 K and V tiles (V transposed into Vt).
    for (int idx = tid; idx < 1024; idx += 128) {
      const int jj = idx >> 4;
      const int d4 = (idx & 15) << 2;
      float4 kq = *(const float4*)(Kp + (size_t)(j0 + jj) * DHEAD + d4);
      *(float4*)(Kt + jj * 68 + d4) = kq;
      float4 vv = *(const float4*)(Vp + (size_t)(j0 + jj) * DHEAD + d4);
      Vt[(d4 + 0) * 68 + jj] = vv.x;
      Vt[(d4 + 1) * 68 + jj] = vv.y;
      Vt[(d4 + 2) * 68 + jj] = vv.z;
      Vt[(d4 + 3) * 68 + jj] = vv.w;
    }
    __syncthreads();

    // S = scale * Q @ K^T   (16 x 64 tile in 4 accumulators)
    v8f s[4] = {zero8, zero8, zero8, zero8};
    for (int kk = 0; kk < 16; ++kk) {
      const int kb = 4 * kk + 2 * half;
      const v2f a = qf[kk];
#pragma unroll
      for (int nb = 0; nb < 4; ++nb) {
        v2f bk = *(const v2f*)(Kt + (nb * 16 + l16) * 68 + kb);
        s[nb] = wmma4(a, bk, s[nb]);
      }
    }

    // += spd bias (this is the once-only 128MB stream)
#pragma unroll
    for (int nb = 0; nb < 4; ++nb)
#pragma unroll
      for (int r = 0; r < 8; ++r)
        s[nb][r] += Sp[(size_t)(r + 8 * half) * SEQ + j0 + nb * 16 + l16];

    if (j0 + 64 < SEQ) {  // hint the next spd block into cache
#pragma unroll
      for (int r = 0; r < 8; ++r)
        __builtin_prefetch(Sp + (size_t)(r + 8 * half) * SEQ + j0 + 64 + l16,
                           0, 0);
    }

    // Row max across the 16 lanes of each half-wave (one row per C-VGPR).
    float pm[8];
#pragma unroll
    for (int r = 0; r < 8; ++r)
      pm[r] = fmaxf(fmaxf(s[0][r], s[1][r]), fmaxf(s[2][r], s[3][r]));
#pragma unroll
    for (int off = 1; off < 16; off <<= 1)
#pragma unroll
      for (int r = 0; r < 8; ++r)
        pm[r] = fmaxf(pm[r], __shfl_xor(pm[r], off, 32));

    float corr[8];
#pragma unroll
    for (int r = 0; r < 8; ++r) {
      const float mn = fmaxf(mrow[r], pm[r]);
      corr[r] = __expf(mrow[r] - mn);
      mrow[r] = mn;
      lrow[r] *= corr[r];
    }
#pragma unroll
    for (int nb = 0; nb < 4; ++nb)
#pragma unroll
      for (int r = 0; r < 8; ++r)
        s[nb][r] = __expf(s[nb][r] - mrow[r]);

    float ps[8];
#pragma unroll
    for (int r = 0; r < 8; ++r)
      ps[r] = (s[0][r] + s[1][r]) + (s[2][r] + s[3][r]);
#pragma unroll
    for (int off = 1; off < 16; off <<= 1)
#pragma unroll
      for (int r = 0; r < 8; ++r)
        ps[r] += __shfl_xor(ps[r], off, 32);
#pragma unroll
    for (int r = 0; r < 8; ++r) lrow[r] += ps[r];

#pragma unroll
    for (int db = 0; db < 4; ++db)
#pragma unroll
      for (int r = 0; r < 8; ++r) o[db][r] *= corr[r];

    // Stage P (C layout -> row-major LDS) then O += P @ V.
#pragma unroll
    for (int nb = 0; nb < 4; ++nb)
#pragma unroll
      for (int r = 0; r < 8; ++r)
        Pl[(r + 8 * half) * 68 + nb * 16 + l16] = s[nb][r];

    asm volatile("s_wait_dscnt 0" ::: "memory");  // same-wave LDS RAW

    for (int kk = 0; kk < 16; ++kk) {
      const int kb = 4 * kk + 2 * half;
      v2f a = *(const v2f*)(Pl + l16 * 68 + kb);
#pragma unroll
      for (int db = 0; db < 4; ++db) {
        v2f bv = *(const v2f*)(Vt + (db * 16 + l16) * 68 + kb);
        o[db] = wmma4(a, bv, o[db]);
      }
    }
  }

  // Finalize: 1/l, head_keep * H / sum(head_keep); store [b][n][h*64+d].
  float ksum = 0.f;
#pragma unroll
  for (int i = 0; i < HEADS; ++i) ksum += head_keep[i];
  const float hk = head_keep[h] * ((float)HEADS / ksum);

  float rs[8];
#pragma unroll
  for (int r = 0; r < 8; ++r) rs[r] = hk / lrow[r];

#pragma unroll
  for (int db = 0; db < 4; ++db)
#pragma unroll
    for (int r = 0; r < 8; ++r) {
      const int row = q0 + r + 8 * half;
      AO[(size_t)(b * SEQ + row) * DIM + h * DHEAD + db * 16 + l16] =
          o[db][r] * rs[r];
    }
}

// ---------------------------------------------------------------------------
// K3: output projection.  A[4096][512] @ W[512][512] + bias -> out.
// ---------------------------------------------------------------------------
__global__ void __launch_bounds__(32) out_proj_kernel(
    const float* __restrict__ A, const float* __restrict__ W,
    const float* __restrict__ bias, float* __restrict__ out) {
  const int lane = threadIdx.x & 31;
  const int half = lane >> 4;
  const int l16  = lane & 15;
  const int m0 = blockIdx.x * 32;
  const int n0 = blockIdx.y * 64;

  const v8f zero8 = {0.f, 0.f, 0.f, 0.f, 0.f, 0.f, 0.f, 0.f};
  v8f acc[2][4];
#pragma unroll
  for (int mi = 0; mi < 2; ++mi)
#pragma unroll
    for (int nb = 0; nb < 4; ++nb) acc[mi][nb] = zero8;

  const float* Arow0 = A + (size_t)(m0 + l16) * DIM;
  const float* Arow1 = A + (size_t)(m0 + 16 + l16) * DIM;

  for (int k0 = 0; k0 < DIM; k0 += 4) {
    const int ka = k0 + 2 * half;
    v2f a0 = *(const v2f*)(Arow0 + ka);
    v2f a1 = *(const v2f*)(Arow1 + ka);
#pragma unroll
    for (int nb = 0; nb < 4; ++nb) {
      const int n = n0 + nb * 16 + l16;
      v2f b;
      b.x = W[(size_t)(ka    ) * DIM + n];
      b.y = W[(size_t)(ka + 1) * DIM + n];
      acc[0][nb] = wmma4(a0, b, acc[0][nb]);
      acc[1][nb] = wmma4(a1, b, acc[1][nb]);
    }
  }

#pragma unroll
  for (int mi = 0; mi < 2; ++mi)
#pragma unroll
    for (int nb = 0; nb < 4; ++nb) {
      const int col = n0 + nb * 16 + l16;
      const float bv = bias[col];
#pragma unroll
      for (int r = 0; r < 8; ++r) {
        const int grow = m0 + mi * 16 + (half << 3) + r;
        out[(size_t)grow * DIM + col] = acc[mi][nb][r] + bv;
      }
    }
}

// ---------------------------------------------------------------------------
extern "C" void kernel_launch(void* const* d_in, const int* in_sizes, int n_in,
                              void* d_out, int out_size, void* d_ws,
                              size_t ws_size, hipStream_t stream) {
  (void)in_sizes; (void)n_in; (void)out_size; (void)ws_size;
  const float* x    = (const float*)d_in[0];  // [4,1024,512]
  const float* spd  = (const float*)d_in[1];  // [4,8,1024,1024]
  const float* hkp  = (const float*)d_in[2];  // [8]
  const float* wqkv = (const float*)d_in[3];  // [512,1536]
  const float* wout = (const float*)d_in[4];  // [512,512]
  const float* bout = (const float*)d_in[5];  // [512]
  float* out = (float*)d_out;                 // [4,1024,512]
  float* ws  = (float*)d_ws;                  // needs 32 MB

  float* q  = ws;
  float* k  = ws + (size_t)QKVSZ;
  float* v  = ws + (size_t)2 * QKVSZ;
  float* ao = ws + (size_t)3 * QKVSZ;

  qkv_gemm_kernel<<<dim3(4096 / 32, NQKV / 64), 32, 0, stream>>>(x, wqkv, ws);
  flash_attn_kernel<<<dim3(SEQ / 64, HEADS, BATCH), 128, 0, stream>>>(
      q, k, v, spd, hkp, ao);
  out_proj_kernel<<<dim3(4096 / 32, DIM / 64), 32, 0, stream>>>(ao, wout, bout,
                                                                out);
}